// GlobalRouter_57483842289992
// MI455X (gfx1250) — compile-verified
//
#include <hip/hip_runtime.h>

typedef __attribute__((ext_vector_type(2))) float v2f;
typedef __attribute__((ext_vector_type(8))) float v8f;

#define HDIM 4096
#define EDIM 64
#define KSPLIT 8
#define CTILES (HDIM / 16)   // 256 column tiles of 16

// ---------------------------------------------------------------------------
// Kernel 0: zero the h-accumulator workspace (4096 floats).
// ---------------------------------------------------------------------------
__global__ void zero_h(float* __restrict__ hacc) {
    int i = blockIdx.x * blockDim.x + threadIdx.x;
    if (i < HDIM) hacc[i] = 0.0f;
}

// ---------------------------------------------------------------------------
// Kernel 1: hacc += x0^T @ W1 via V_WMMA_F32_16X16X4_F32 (GEMV-as-GEMM).
// One wave owns one 16-column tile and one K-slice of 512; A has only row 0
// populated. Wave id is readfirstlane'd so loop bounds / prefetch guard are
// SGPRs -> uniform s_cmp loop, EXEC stays all-ones through every v_wmma.
// 2048 waves total = 256 blocks x 8 waves (wave32).
// ---------------------------------------------------------------------------
__global__ __launch_bounds__(256) void gemv1_wmma(const float* __restrict__ x0,
                                                  const float* __restrict__ W1,
                                                  float* __restrict__ hacc) {
    const int lane = threadIdx.x & 31;
    // Scalarize the wave index: identical across the wave by construction.
    const int wave = __builtin_amdgcn_readfirstlane(threadIdx.x >> 5);
    const int gw   = blockIdx.x * 8 + wave;         // 0..2047 (scalar)
    const int tile = gw & (CTILES - 1);             // column tile (scalar)
    const int kslc = gw >> 8;                       // 0..KSPLIT-1 (scalar)
    const int j0   = tile * 16;
    const int kbeg = kslc * (HDIM / KSPLIT);
    const int kend = kbeg + (HDIM / KSPLIT);

    const int half = lane >> 4;                     // 0 -> K{0,1}, 1 -> K{2,3}
    const int lm   = lane & 15;                     // A: M row / B: N column
    const float am = (lm == 0) ? 1.0f : 0.0f;       // hoisted row-0 mask for A

    // Per-lane streaming bases (advance by immediate-friendly strides).
    const float* xp = x0 + kbeg + 2 * half;                          // A source
    const float* wp = W1 + (size_t)(kbeg + 2 * half) * HDIM + j0 + lm; // B source

    v8f d = {0.f, 0.f, 0.f, 0.f, 0.f, 0.f, 0.f, 0.f};

    #pragma unroll 4
    for (int k0 = kbeg; k0 < kend; k0 += 4) {
        // A-matrix 16x4 f32: only row M=0 live (lane 0: K0,K1; lane 16: K2,K3)
        v2f a;
        a.x = xp[0] * am;
        a.y = xp[1] * am;

        // B-matrix 4x16 f32 tile of W1: lane = column N, half-wave = K pair.
        v2f b;
        b.x = wp[0];            // row kk
        b.y = wp[HDIM];         // row kk+1

        // Scalar-guarded prefetch 4 k-steps (16 rows) ahead, WGP scope.
        if (k0 + 16 < kend)
            __builtin_prefetch((const void*)(wp + (size_t)16 * HDIM), 0, 3);

        // D = A*B + D  ->  v_wmma_f32_16x16x4_f32
        d = __builtin_amdgcn_wmma_f32_16x16x4_f32(
                /*neg_a=*/false, a, /*neg_b=*/false, b,
                /*c_mod=*/(short)0, d, /*reuse_a=*/false, /*reuse_b=*/false);

        xp += 4;
        wp += (size_t)4 * HDIM;
    }

    // Row M=0 of D lives in VGPR0 (d[0]) of lanes 0..15, N = lane.
    if (lane < 16)
        atomicAdd(&hacc[j0 + lane], d[0]);
}

// ---------------------------------------------------------------------------
// Kernel 2: act = relu(hacc + b1) (staged in LDS); logits = act @ W2 + b2;
// top-2 + softmax; scatter 64 outputs. Single 256-thread block.
// ---------------------------------------------------------------------------
__global__ __launch_bounds__(256) void finish(const float* __restrict__ hacc,
                                              const float* __restrict__ b1,
                                              const float* __restrict__ W2,
                                              const float* __restrict__ b2,
                                              float* __restrict__ out) {
    __shared__ float act[HDIM];          // 16 KB of the 320 KB WGP LDS
    __shared__ float part[4][EDIM];
    __shared__ float logits[EDIM];
    __shared__ float res[EDIM];

    const int tid = threadIdx.x;

    for (int k = tid; k < HDIM; k += 256)
        act[k] = fmaxf(hacc[k] + b1[k], 0.0f);
    __syncthreads();

    const int j     = tid & 63;          // output expert
    const int slice = tid >> 6;          // 0..3 k-slices of 1024
    float acc = 0.0f;
    const int ks = slice * (HDIM / 4);
    const int ke = ks + (HDIM / 4);
    for (int k = ks; k < ke; ++k)
        acc += act[k] * W2[(size_t)k * EDIM + j];
    part[slice][j] = acc;
    __syncthreads();

    if (tid < EDIM)
        logits[tid] = part[0][tid] + part[1][tid] + part[2][tid] + part[3][tid]
                    + b2[tid];
    __syncthreads();

    if (tid == 0) {
        // top-2 (first-occurrence tie-break like jax.lax.top_k)
        float v1 = -1e30f; int i1 = 0;
        for (int n = 0; n < EDIM; ++n) {
            float v = logits[n];
            if (v > v1) { v1 = v; i1 = n; }
        }
        float v2 = -1e30f; int i2 = 0;
        for (int n = 0; n < EDIM; ++n) {
            if (n == i1) continue;
            float v = logits[n];
            if (v > v2) { v2 = v; i2 = n; }
        }
        // softmax over {v1, v2}; TEMPERATURE == 1.0
        const float e2 = expf(v2 - v1);
        const float p1 = 1.0f / (1.0f + e2);
        const float p2 = e2 * p1;
        for (int n = 0; n < EDIM; ++n) res[n] = 0.0f;
        res[i1] = p1;
        res[i2] = p2;
    }
    __syncthreads();

    if (tid < EDIM) out[tid] = res[tid];
}

// ---------------------------------------------------------------------------
extern "C" void kernel_launch(void* const* d_in, const int* in_sizes, int n_in,
                              void* d_out, int out_size, void* d_ws, size_t ws_size,
                              hipStream_t stream) {
    (void)in_sizes; (void)n_in; (void)out_size; (void)ws_size;
    const float* x  = (const float*)d_in[0];  // (4096, 8, 4096); x[0,0,:] = first 4096
    const float* W1 = (const float*)d_in[1];  // (4096, 4096) row-major
    const float* b1 = (const float*)d_in[2];  // (4096)
    const float* W2 = (const float*)d_in[3];  // (4096, 64) row-major
    const float* b2 = (const float*)d_in[4];  // (64)
    // d_in[5] = top_k (== 2, baked into finish kernel)

    float* hacc = (float*)d_ws;               // 4096 floats of scratch

    zero_h<<<HDIM / 256, 256, 0, stream>>>(hacc);
    gemv1_wmma<<<(CTILES * KSPLIT) / 8, 256, 0, stream>>>(x, W1, hacc);
    finish<<<1, 256, 0, stream>>>(hacc, b1, W2, b2, (float*)d_out);
}